// ImageLoss_79955111182639
// MI455X (gfx1250) — compile-verified
//
#include <hip/hip_runtime.h>
#include <hip/hip_bf16.h>
#include <stdint.h>

#define NPIX 4096          // 64x64 grid
#define NB   8             // batch
#define NTOT (NB * NPIX)   // 32768
#define BIGF 1.0e6f

typedef __attribute__((ext_vector_type(2))) float v2f;
typedef __attribute__((ext_vector_type(8))) float v8f;

// ---------------------------------------------------------------------------
// Kernel 1: per-batch min scans.  grid = (8 strips, 8 batches), 256 threads.
// Stages pred[b], gt[b] into LDS via async-to-LDS (ASYNCcnt), builds the
// |dh|,|dw| -> sqrt(dh^2+dw^2)+1 LUT in LDS, then each thread scans all 4096
// (i,j) for 2 output pixels, producing gt_th*min and pred_th*min contributions.
// ---------------------------------------------------------------------------
__global__ __launch_bounds__(256)
void minscan_kernel(const float* __restrict__ gt, const float* __restrict__ pred,
                    float* __restrict__ contrib_p, float* __restrict__ contrib_g) {
  __shared__ float pred_s[NPIX];
  __shared__ float gt_s[NPIX];
  __shared__ float dtab[NPIX];

  const int tid   = threadIdx.x;
  const int strip = blockIdx.x;   // 0..7 -> 512 output pixels per block
  const int b     = blockIdx.y;   // 0..7 batch

  // ---- async stage pred[b], gt[b] into LDS (16 KB each, b128 per lane) ----
  {
    unsigned lds_pred = (unsigned)(uintptr_t)&pred_s[0];
    unsigned lds_gt   = (unsigned)(uintptr_t)&gt_s[0];
    unsigned long long gp = (unsigned long long)(uintptr_t)(pred + b * NPIX);
    unsigned long long gg = (unsigned long long)(uintptr_t)(gt   + b * NPIX);
#pragma unroll
    for (int k = 0; k < 4; ++k) {
      unsigned off = (unsigned)(tid + k * 256) * 16u;
      unsigned lp = lds_pred + off;
      unsigned lg = lds_gt + off;
      asm volatile("global_load_async_to_lds_b128 %0, %1, %2"
                   :: "v"(lp), "v"(off), "s"(gp) : "memory");
      asm volatile("global_load_async_to_lds_b128 %0, %1, %2"
                   :: "v"(lg), "v"(off), "s"(gg) : "memory");
    }
  }

  // ---- distance LUT: dtab[dh*64+dw] = sqrt(dh^2+dw^2) + 1 ----
  for (int t = tid; t < NPIX; t += 256) {
    int dh = t >> 6, dw = t & 63;
    dtab[t] = __builtin_sqrtf((float)(dh * dh + dw * dw)) + 1.0f;
  }

  asm volatile("s_wait_asynccnt 0x0" ::: "memory");
  __syncthreads();

  // ---- min scans: 2 output pixels per thread, full 4096-point inner scan ----
#pragma unroll
  for (int k = 0; k < 2; ++k) {
    const int p = strip * 512 + k * 256 + tid;
    const int h = p >> 6, w = p & 63;
    float minp = 3.0e38f, ming = 3.0e38f;
    for (int i = 0; i < 64; ++i) {
      int dh = h - i; dh = dh < 0 ? -dh : dh;
      const float* __restrict__ drow = &dtab[dh << 6];
      const float* __restrict__ pr   = &pred_s[i << 6];
      const float* __restrict__ gr   = &gt_s[i << 6];
#pragma unroll 8
      for (int j = 0; j < 64; ++j) {
        int dw = w - j; dw = dw < 0 ? -dw : dw;
        float d = drow[dw];
        minp = fminf(minp, d * pr[j]);
        ming = fminf(ming, d * gr[j]);
      }
    }
    const float gtv = gt_s[p], prv = pred_s[p];
    const float gt_th = gtv + (1.0f - gtv) * BIGF;  // > 0 always, hoists from min
    const float pr_th = prv + (1.0f - prv) * BIGF;
    contrib_p[b * NPIX + p] = gt_th * minp;
    contrib_g[b * NPIX + p] = pr_th * ming;
  }
}

// ---------------------------------------------------------------------------
// Kernel 2: WMMA-based sum reductions. 1 block, 8 wave32s.
// v_wmma_f32_16x16x4_f32 with all-ones B: sum of all 256 D entries == 16 * sum
// of all 64 A entries, regardless of the exact A/B/D lane layouts (ones matrix
// and a full-sum are layout-invariant) -> robust 64-elem/instr fp32 reducer.
// ---------------------------------------------------------------------------
__global__ __launch_bounds__(256)
void reduce_kernel(const float* __restrict__ gt, const float* __restrict__ pred,
                   const float* __restrict__ cp, const float* __restrict__ cg,
                   float* __restrict__ out) {
  __shared__ float part[3][8];
  const int tid  = threadIdx.x;
  const int wave = tid >> 5;
  const int lane = tid & 31;

  v8f accs = {};  // sum of (gt-pred)^2
  v8f accp = {};  // sum of gt_th * min(dist*pred)
  v8f accg = {};  // sum of pred_th * min(dist*gt)
  v2f ones; ones.x = 1.0f; ones.y = 1.0f;

  const int base = wave * (NTOT / 8) + lane * 2;  // 4096 elements per wave
  for (int t = 0; t < 64; ++t) {
    const int idx = base + t * 64;
    float d0 = gt[idx]     - pred[idx];
    float d1 = gt[idx + 1] - pred[idx + 1];
    v2f a;  a.x = d0 * d0;   a.y = d1 * d1;
    v2f ap; ap.x = cp[idx];  ap.y = cp[idx + 1];
    v2f ag; ag.x = cg[idx];  ag.y = cg[idx + 1];
    accs = __builtin_amdgcn_wmma_f32_16x16x4_f32(false, a,  false, ones, (short)0, accs, false, false);
    accp = __builtin_amdgcn_wmma_f32_16x16x4_f32(false, ap, false, ones, (short)0, accp, false, false);
    accg = __builtin_amdgcn_wmma_f32_16x16x4_f32(false, ag, false, ones, (short)0, accg, false, false);
  }

  float ss = 0.0f, sp = 0.0f, sg = 0.0f;
#pragma unroll
  for (int c = 0; c < 8; ++c) { ss += accs[c]; sp += accp[c]; sg += accg[c]; }
#pragma unroll
  for (int m = 16; m >= 1; m >>= 1) {
    ss += __shfl_xor(ss, m, 32);
    sp += __shfl_xor(sp, m, 32);
    sg += __shfl_xor(sg, m, 32);
  }
  if (lane == 0) { part[0][wave] = ss; part[1][wave] = sp; part[2][wave] = sg; }
  __syncthreads();

  if (tid == 0) {
    float ts = 0.0f, tp = 0.0f, tg = 0.0f;
#pragma unroll
    for (int wv = 0; wv < 8; ++wv) { ts += part[0][wv]; tp += part[1][wv]; tg += part[2][wv]; }
    ts *= (1.0f / 16.0f);  // undo 16x overcount from summing all D entries
    tp *= (1.0f / 16.0f);
    tg *= (1.0f / 16.0f);
    out[0] = ts / 512.0f;       // mean over (B,H) of W-sums
    out[1] = tp / 32768.0f;     // mean over (B,H,W)
    out[2] = tg / 32768.0f;
  }
}

extern "C" void kernel_launch(void* const* d_in, const int* in_sizes, int n_in,
                              void* d_out, int out_size, void* d_ws, size_t ws_size,
                              hipStream_t stream) {
  const float* gt   = (const float*)d_in[0];
  const float* pred = (const float*)d_in[1];
  float* wsf = (float*)d_ws;
  float* contrib_p = wsf;            // NTOT floats
  float* contrib_g = wsf + NTOT;     // NTOT floats (256 KB total scratch use)

  minscan_kernel<<<dim3(8, 8), 256, 0, stream>>>(gt, pred, contrib_p, contrib_g);
  reduce_kernel<<<1, 256, 0, stream>>>(gt, pred, contrib_p, contrib_g, (float*)d_out);
}